// Network_44452911514061
// MI455X (gfx1250) — compile-verified
//
#include <hip/hip_runtime.h>
#include <math.h>

#define F       8
#define HH      1024
#define WW      1024
#define NPTS    65536
#define TITER   48
#define WARMUP  8
#define HW      (HH * WW)

typedef float v2f __attribute__((ext_vector_type(2)));
typedef float v8f __attribute__((ext_vector_type(8)));

// single-instruction hardware reciprocal (v_rcp_f32) -- image-precision OK
__device__ __forceinline__ float fast_rcp(float x) {
    return __builtin_amdgcn_rcpf(x);
}

// ---------------------------------------------------------------------------
// Counter-based threefry2x32 (deterministic, stateless RNG)
// ---------------------------------------------------------------------------
__device__ __forceinline__ uint2 threefry2x32(uint2 k, uint2 c) {
    const unsigned R[8] = {13u, 15u, 26u, 6u, 17u, 29u, 16u, 24u};
    unsigned ks[3];
    ks[0] = k.x; ks[1] = k.y; ks[2] = 0x1BD11BDAu ^ k.x ^ k.y;
    unsigned x0 = c.x + ks[0];
    unsigned x1 = c.y + ks[1];
#pragma unroll
    for (int r = 0; r < 20; ++r) {
        x0 += x1;
        unsigned rot = R[r & 7];
        x1 = (x1 << rot) | (x1 >> (32u - rot));
        x1 ^= x0;
        if ((r & 3) == 3) {
            unsigned s = (unsigned)(r >> 2) + 1u;
            x0 += ks[s % 3u];
            x1 += ks[(s + 1u) % 3u] + s;
        }
    }
    uint2 out; out.x = x0; out.y = x1;
    return out;
}

__device__ __forceinline__ float u01(unsigned v) {
    return (float)(v >> 8) * 0x1p-24f;   // [0,1)
}

// fast sigmoid: v_rcp + v_exp, no IEEE div ladder
__device__ __forceinline__ float sigmoidf(float x) {
    return fast_rcp(1.0f + __expf(-x));
}

// ---------------------------------------------------------------------------
// Zero the F x H x W histogram (must run every launch: atomics accumulate)
// ---------------------------------------------------------------------------
__global__ void zero_hist_kernel(float4* __restrict__ p, int n4) {
    int i = blockIdx.x * blockDim.x + threadIdx.x;
    if (i < n4) p[i] = make_float4(0.f, 0.f, 0.f, 0.f);
}

// ---------------------------------------------------------------------------
// One chaos-game step: inverse-CDF (two ds_load_b128), affine, bilinear splat
// ---------------------------------------------------------------------------
__device__ __forceinline__ void chaos_step(float& x0, float& x1, int& s,
                                           float u, int t,
                                           const float* __restrict__ sCumT,
                                           const float* __restrict__ sAff,
                                           float* __restrict__ hist)
{
    const float4* col = (const float4*)&sCumT[s * F];
    float4 c0 = col[0];
    float4 c1 = col[1];
    int nxt = (c0.x < u) + (c0.y < u) + (c0.z < u) + (c0.w < u)
            + (c1.x < u) + (c1.y < u) + (c1.z < u) + (c1.w < u);
    nxt = min(nxt, F - 1);

    const float4 a03 = *(const float4*)&sAff[nxt * 8];      // A00 A01 A02 A10
    const float2 a45 = *(const float2*)&sAff[nxt * 8 + 4];  // A11 A12
    float nx = fmaf(a03.x, x0, fmaf(a03.y, x1, a03.z));
    float ny = fmaf(a03.w, x0, fmaf(a45.x, x1, a45.y));
    x0 = nx; x1 = ny; s = nxt;

    if (t >= WARMUP) {
        float gx = (x0 + 1.f) * 0.5f * (float)WW - 0.5f;
        float gy = (x1 + 1.f) * 0.5f * (float)HH - 0.5f;
        float fx0 = floorf(gx), fy0 = floorf(gy);
        float fx = gx - fx0,   fy = gy - fy0;
        int xi = (int)fx0, yi = (int)fy0;
        float* plane = hist + (size_t)nxt * HW;
        const float wts[4] = {(1.f - fx) * (1.f - fy), fx * (1.f - fy),
                              (1.f - fx) * fy,         fx * fy};
        const int dxs[4] = {0, 1, 0, 1};
        const int dys[4] = {0, 0, 1, 1};
#pragma unroll
        for (int k = 0; k < 4; ++k) {
            int xx = xi + dxs[k], yy = yi + dys[k];
            if (xx >= 0 && xx < WW && yy >= 0 && yy < HH)
                unsafeAtomicAdd(plane + (size_t)yy * WW + xx, wts[k]);
        }
    }
}

// ---------------------------------------------------------------------------
// Chaos game: one thread per point, T iterations. One threefry draw feeds
// TWO iterations (.x then .y) -- halves the dominant RNG ALU cost.
// ---------------------------------------------------------------------------
__global__ void __launch_bounds__(256)
chaos_kernel(const float* __restrict__ prob,     // (F,F) raw logits
             const float* __restrict__ affine,   // (F,2,3)
             float*       __restrict__ hist)     // (F,H,W)
{
    // sCumT[cur][next]: column 'cur' of cumP stored contiguously (8 floats)
    __shared__ __align__(16) float sCumT[F * F];
    __shared__ __align__(16) float sAff[F * 8];   // padded to 8 floats/func

    const int tid = threadIdx.x;

    if (tid < F) {  // thread j builds column j of column-stochastic cumsum
        float p[F]; float sum = 0.f;
#pragma unroll
        for (int i = 0; i < F; ++i) {
            float v = sigmoidf(prob[i * F + tid]) + 0.001f;
            p[i] = v; sum += v;
        }
        float acc = 0.f;
        const float inv = fast_rcp(sum);
#pragma unroll
        for (int i = 0; i < F; ++i) {
            acc += p[i] * inv;
            sCumT[tid * F + i] = acc;          // transposed: column-contiguous
        }
    }
    if (tid < F * 6) {
        int f = tid / 6, k = tid - f * 6;
        sAff[f * 8 + k] = affine[tid];
    }
    __syncthreads();

    const int pid = blockIdx.x * blockDim.x + tid;

    // initial position in [-1,1)^2 and initial state
    uint2 ka; ka.x = 0x9E3779B9u; ka.y = 42u;
    uint2 ca; ca.x = (unsigned)pid; ca.y = 0xA5A5A5A5u;
    uint2 r0 = threefry2x32(ka, ca);
    float x0 = u01(r0.x) * 2.f - 1.f;
    float x1 = u01(r0.y) * 2.f - 1.f;
    uint2 kb; kb.x = 0x85EBCA6Bu; kb.y = 7u;
    uint2 cb; cb.x = (unsigned)pid; cb.y = 0x5A5A5A5Au;
    int s = (int)(threefry2x32(kb, cb).x & (F - 1u));

    uint2 kt; kt.x = 0xC2B2AE35u; kt.y = 0x27220A95u;
    for (int t = 0; t < TITER; t += 2) {
        uint2 ct; ct.x = (unsigned)pid; ct.y = (unsigned)(t >> 1);
        uint2 rr = threefry2x32(kt, ct);
        chaos_step(x0, x1, s, u01(rr.x), t,     sCumT, sAff, hist);
        chaos_step(x0, x1, s, u01(rr.y), t + 1, sCumT, sAff, hist);
    }
}

// ---------------------------------------------------------------------------
// Colouring: rgb = CT(3x8) x colour_n(8xHW) via V_WMMA_F32_16X16X4_F32
// (two chained K=4 steps), then log/pow tone-map + background blend.
// One wave handles 16 pixels per WMMA tile; lanes L and 16+L share pixel L.
// Fast transcendentals throughout: the pass is HBM-bound.
// ---------------------------------------------------------------------------
__global__ void __launch_bounds__(256)
colour_kernel(const float* __restrict__ hist,  // (F,H,W)
              const float* __restrict__ ctp,   // (3,F) raw logits
              const float* __restrict__ bgp,   // (3,)  raw logits
              const float* __restrict__ expP,
              const float* __restrict__ krP,
              const float* __restrict__ gmP,
              float*       __restrict__ out)   // (3,H,W)
{
    const int tid   = threadIdx.x;
    const int lane  = tid & 31;
    const int wave  = tid >> 5;
    const int L     = lane & 15;        // pixel slot / A-matrix row
    const bool hi   = (lane >> 4) != 0; // which K-half this lane holds
    const int tile  = blockIdx.x * (blockDim.x >> 5) + wave;
    const int pix   = tile * 16 + L;

    // per-pixel histogram vector + sum (lanes L and 16+L load the same pixel)
    float h0 = hist[(size_t)0 * HW + pix];
    float h1 = hist[(size_t)1 * HW + pix];
    float h2 = hist[(size_t)2 * HW + pix];
    float h3 = hist[(size_t)3 * HW + pix];
    float h4 = hist[(size_t)4 * HW + pix];
    float h5 = hist[(size_t)5 * HW + pix];
    float h6 = hist[(size_t)6 * HW + pix];
    float h7 = hist[(size_t)7 * HW + pix];
    const float hs = ((h0 + h1) + (h2 + h3)) + ((h4 + h5) + (h6 + h7));
    const float inv = fast_rcp(hs + 1e-5f);

    const float n0 = h0 * inv, n1 = h1 * inv, n2 = h2 * inv, n3 = h3 * inv;
    const float n4 = h4 * inv, n5 = h5 * inv, n6 = h6 * inv, n7 = h7 * inv;

    // A = sigmoid(CT) padded to 16 rows of zeros (rows >= 3)
    const bool row = (L < 3);
    const float t0 = row ? sigmoidf(ctp[L * F + 0]) : 0.f;
    const float t1 = row ? sigmoidf(ctp[L * F + 1]) : 0.f;
    const float t2 = row ? sigmoidf(ctp[L * F + 2]) : 0.f;
    const float t3 = row ? sigmoidf(ctp[L * F + 3]) : 0.f;
    const float t4 = row ? sigmoidf(ctp[L * F + 4]) : 0.f;
    const float t5 = row ? sigmoidf(ctp[L * F + 5]) : 0.f;
    const float t6 = row ? sigmoidf(ctp[L * F + 6]) : 0.f;
    const float t7 = row ? sigmoidf(ctp[L * F + 7]) : 0.f;

    // Operand layouts per ISA (16x4 f32 A, 4x16 B): single cndmask per pick
    v2f a0, a1, b0, b1;
    a0.x = hi ? t2 : t0;   a0.y = hi ? t3 : t1;   // A: K = 2*hi, 2*hi+1
    a1.x = hi ? t6 : t4;   a1.y = hi ? t7 : t5;
    b0.x = hi ? n1 : n0;   b0.y = hi ? n3 : n2;   // B: K = hi, 2+hi
    b1.x = hi ? n5 : n4;   b1.y = hi ? n7 : n6;

    v8f c = {};
    c = __builtin_amdgcn_wmma_f32_16x16x4_f32(false, a0, false, b0,
                                              (short)0, c, false, false);
    c = __builtin_amdgcn_wmma_f32_16x16x4_f32(false, a1, false, b1,
                                              (short)0, c, false, false);

    // tone map (fast log: alpha is a clamped blend factor, v_log_f32 is ample)
    const float ex = expP[0], kr = krP[0], gm = gmP[0];
    const float baseline = (float)HW / (float)(NPTS * (TITER - WARMUP));
    const float nexp = ex * baseline * fast_rcp(kr * kr);
    const float intensity = __logf(1.0f + hs * nexp);
    const float alpha = fminf(fmaxf(intensity, 0.f), 1.f);
    const float invg = fast_rcp(gm);

    if (lane < 16) {   // lanes 0..15 hold D rows M=0..2 -> channels 0..2
#pragma unroll
        for (int ch = 0; ch < 3; ++ch) {
            float rgb = c[ch];
            float img = __powf(fmaxf(rgb * alpha, 1e-8f), invg);
            float bg  = sigmoidf(bgp[ch]);
            out[(size_t)ch * HW + pix] = img + bg * (1.f - alpha);
        }
    }
}

// ---------------------------------------------------------------------------
extern "C" void kernel_launch(void* const* d_in, const int* in_sizes, int n_in,
                              void* d_out, int out_size, void* d_ws, size_t ws_size,
                              hipStream_t stream)
{
    const float* prob = (const float*)d_in[0];   // (1,F,F)
    const float* aff  = (const float*)d_in[1];   // (1,F,2,3)
    const float* ct   = (const float*)d_in[2];   // (1,3,F)
    const float* bg   = (const float*)d_in[3];   // (1,3)
    const float* ex   = (const float*)d_in[4];
    const float* kr   = (const float*)d_in[5];
    const float* gm   = (const float*)d_in[6];
    float* out  = (float*)d_out;
    float* hist = (float*)d_ws;                  // F*H*W floats = 32 MB

    const int n4 = (F * HW) / 4;
    zero_hist_kernel<<<(n4 + 255) / 256, 256, 0, stream>>>((float4*)hist, n4);
    chaos_kernel<<<NPTS / 256, 256, 0, stream>>>(prob, aff, hist);
    // HW/16 wave-tiles, 8 waves per 256-thread block
    colour_kernel<<<(HW / 16) / 8, 256, 0, stream>>>(hist, ct, bg, ex, kr, gm, out);
}